// IDWT_5540507811879
// MI455X (gfx1250) — compile-verified
//
#include <hip/hip_runtime.h>

// IDWT-Haar reconstruction, fused closed form:
//   out[b, 2m+dr, 2n+dc] = 0.5*( x0 + x1 + sc*x2 + sr*sc*x3 ),
//   x_k = in[b, m, n, k],  sr = dr?+1:-1, sc = dc?+1:-1
// Pure streaming gather/scatter: 64MB in + 64MB out -> ~5.8us roofline @23.3TB/s.
// CDNA5 path: GLOBAL_LOAD_ASYNC_TO_LDS_B128 double-buffered pipeline + S_WAIT_ASYNCCNT,
// NT b128 stores. No barriers needed (each lane consumes only its own LDS slot).

typedef float v4f __attribute__((ext_vector_type(4)));

#define BLOCK 256
#define GRID  2048
#define TPB   4   // tiles per block; GRID*TPB*BLOCK = 2,097,152 pixel-pairs = all of them

__device__ __forceinline__ void async_load_pair(unsigned lds_off, const float* gptr) {
    // Copies 32 bytes (2 input pixels, 2 x b128) global -> LDS via the async engine.
    // INST_OFFSET is added to BOTH the LDS and the global address (ISA 08 §4.4),
    // so one offset:16 covers the second half on both sides.
    unsigned long long ga = (unsigned long long)gptr;
    asm volatile(
        "global_load_async_to_lds_b128 %0, %1, off\n\t"
        "global_load_async_to_lds_b128 %0, %1, off offset:16"
        :
        : "v"(lds_off), "v"(ga)
        : "memory");
}

__global__ __launch_bounds__(BLOCK) void idwt_haar_kernel(const float* __restrict__ in,
                                                          float* __restrict__ out) {
    __shared__ v4f lds[2][BLOCK][2];   // 2 stages x 256 threads x 32B = 16 KB

    const unsigned tid = threadIdx.x;
    const unsigned lds_base = (unsigned)(unsigned long long)(void*)&lds[0][0][0];
    const unsigned lds_s0 = lds_base + tid * 32u;
    const unsigned lds_s1 = lds_s0 + (unsigned)(BLOCK * 32);

    const unsigned blk = blockIdx.x;

    // Prime the pipeline: tile 0 -> stage 0.
    {
        const unsigned P0 = blk * BLOCK + tid;
        async_load_pair(lds_s0, in + (size_t)P0 * 8u);
    }

#pragma unroll
    for (int k = 0; k < TPB; ++k) {
        if (k + 1 < TPB) {
            // Prefetch next tile into the other stage, then wait until only those
            // 2 async ops remain outstanding (async loads complete in order).
            const unsigned Pn = (blk + (unsigned)(k + 1) * GRID) * BLOCK + tid;
            async_load_pair(((k + 1) & 1) ? lds_s1 : lds_s0, in + (size_t)Pn * 8u);
            asm volatile("s_wait_asynccnt 0x2" ::: "memory");
        } else {
            asm volatile("s_wait_asynccnt 0x0" ::: "memory");
        }

        const unsigned P = (blk + (unsigned)k * GRID) * BLOCK + tid;
        const v4f p0 = lds[k & 1][tid][0];   // pixel 2n   : {x0,x1,x2,x3}
        const v4f p1 = lds[k & 1][tid][1];   // pixel 2n+1 : {x0,x1,x2,x3}

        const float a0 = p0.x + p0.y, e0 = a0 - p0.z, o0 = a0 + p0.z, w0 = p0.w;
        const float a1 = p1.x + p1.y, e1 = a1 - p1.z, o1 = a1 + p1.z, w1 = p1.w;

        v4f topv = { e0 + w0, o0 - w0, e1 + w1, o1 - w1 };   // row 2m
        v4f botv = { e0 - w0, o0 + w0, e1 - w1, o1 + w1 };   // row 2m+1
        topv *= 0.5f;
        botv *= 0.5f;

        // P = ((b*512 + m)*256 + n_pair); input bytes consumed linearly at 32*P.
        const unsigned n = P & 255u;            // pair index in row  -> cols 4n..4n+3
        const unsigned m = (P >> 8) & 511u;     // input row          -> rows 2m, 2m+1
        const unsigned b = P >> 17;             // batch

        const size_t top = ((size_t)(b * 1024u + 2u * m)) * 1024u + 4u * n;
        __builtin_nontemporal_store(topv, (v4f*)(out + top));
        __builtin_nontemporal_store(botv, (v4f*)(out + top + 1024u));
    }
}

extern "C" void kernel_launch(void* const* d_in, const int* in_sizes, int n_in,
                              void* d_out, int out_size, void* d_ws, size_t ws_size,
                              hipStream_t stream) {
    (void)in_sizes; (void)n_in; (void)out_size; (void)d_ws; (void)ws_size;
    const float* in = (const float*)d_in[0];   // (16, 512, 512, 4) f32
    float* out = (float*)d_out;                // (16, 1024, 1024, 1) f32
    idwt_haar_kernel<<<GRID, BLOCK, 0, stream>>>(in, out);
}